// MultiGraphSage_LSTM_70995809403001
// MI455X (gfx1250) — compile-verified
//
#include <hip/hip_runtime.h>
#include <hip/hip_bf16.h>
#include <math.h>

// ---- problem constants (match reference) ----
#define SEQ   4
#define GNUM  3
#define SG    (SEQ*GNUM)       // 12 (s,g) pairs
#define NNODES 20000
#define EDGES  320000
#define DIM    64              // DIN == HID == OUT
#define BATCH  (SEQ*NNODES)    // 80000 LSTM rows

typedef float v2f __attribute__((ext_vector_type(2)));
typedef float v8f __attribute__((ext_vector_type(8)));

// fp32 WMMA: D(16x16,f32) = A(16x4,f32) * B(4x16,f32) + C
__device__ __forceinline__ v8f wmma_f32(v2f a, v2f b, v8f c) {
    return __builtin_amdgcn_wmma_f32_16x16x4_f32(
        /*neg_a=*/false, a, /*neg_b=*/false, b,
        /*c_mod=*/(short)0, c, /*reuse_a=*/false, /*reuse_b=*/false);
}

__device__ __forceinline__ float sigmoidf_(float x) {
    return 1.0f / (1.0f + __expf(-x));
}

// ---------------------------------------------------------------------------
// 0) zero workspace region
// ---------------------------------------------------------------------------
__global__ void zero_kernel(float* __restrict__ p, long long n) {
    long long i = (long long)blockIdx.x * blockDim.x + threadIdx.x;
    if (i < n) p[i] = 0.0f;
}

// ---------------------------------------------------------------------------
// 1) edge scatter: agg[sg][dst] += feat[sg][src]; cnt[sg][dst] += 1
//    one wave per edge, 2 dims per lane -> global_atomic_add_f32
// ---------------------------------------------------------------------------
__global__ void scatter_kernel(const float* __restrict__ feat,  // [SG][N][64]
                               const int*   __restrict__ ei,    // [SG][2][E]
                               float* __restrict__ agg,         // [SG][N][64]
                               float* __restrict__ cnt)         // [SG][N]
{
    const int sg   = blockIdx.y;
    const int lane = threadIdx.x & 31;
    const int e    = blockIdx.x * (blockDim.x >> 5) + (threadIdx.x >> 5);
    if (e >= EDGES) return;

    const int* ei_sg = ei + (size_t)sg * 2 * EDGES;
    const int src = ei_sg[e];
    const int dst = ei_sg[EDGES + e];

    const float* frow = feat + ((size_t)sg * NNODES + src) * DIM;
    float*       arow = agg  + ((size_t)sg * NNODES + dst) * DIM;

    atomicAdd(&arow[lane],      frow[lane]);
    atomicAdd(&arow[lane + 32], frow[lane + 32]);
    if (lane == 0) atomicAdd(&cnt[(size_t)sg * NNODES + dst], 1.0f);
}

// ---------------------------------------------------------------------------
// 2) SAGE linear: out = tanh( (agg/cnt) @ Wl + xin @ Wr + b )
//    256 threads = 8 waves; each wave owns a 32-row x 64-col output stripe
//    (two 16-row WMMA tiles sharing every B-fragment -> half the LDS traffic
//    per WMMA). Weights staged in LDS (2 x 16KB). fp32 WMMA 16x16x4.
// ---------------------------------------------------------------------------
__global__ void sage_linear_kernel(const float* __restrict__ xin,  // [SG][N][64]
                                   const float* __restrict__ agg,  // [SG][N][64]
                                   const float* __restrict__ cnt,  // [SG][N]
                                   const float* __restrict__ Wl,   // [G][64][64]
                                   const float* __restrict__ Wr,   // [G][64][64]
                                   const float* __restrict__ bias, // [G][64]
                                   float* __restrict__ out)        // [SG][N][64]
{
    __shared__ float sWl[DIM * DIM];
    __shared__ float sWr[DIM * DIM];
    __shared__ float sB[DIM];

    const int sg = blockIdx.y;
    const int g  = sg % GNUM;

    for (int i = threadIdx.x; i < DIM * DIM; i += blockDim.x) {
        sWl[i] = Wl[(size_t)g * DIM * DIM + i];
        sWr[i] = Wr[(size_t)g * DIM * DIM + i];
    }
    if (threadIdx.x < DIM) sB[threadIdx.x] = bias[g * DIM + threadIdx.x];
    __syncthreads();

    const int lane = threadIdx.x & 31;
    const int wave = threadIdx.x >> 5;
    const int lo   = lane & 15;
    const int hi   = lane >> 4;

    const int row0 = (blockIdx.x * 8 + wave) * 32;   // wave-uniform, 2 tiles
    if (row0 >= NNODES) return;

    const float* xin_sg = xin + (size_t)sg * NNODES * DIM;
    const float* agg_sg = agg + (size_t)sg * NNODES * DIM;
    const float* cnt_sg = cnt + (size_t)sg * NNODES;
    float*       out_sg = out + (size_t)sg * NNODES * DIM;

    // A rows for the two tiles (clamped so OOB lanes read something valid)
    const int  ar0   = row0 + lo;
    const int  ar1   = row0 + 16 + lo;
    const int  ar0c  = (ar0 < NNODES) ? ar0 : (NNODES - 1);
    const int  ar1c  = (ar1 < NNODES) ? ar1 : (NNODES - 1);

    const float c00   = cnt_sg[ar0c];
    const float c01   = cnt_sg[ar1c];
    const float scl0  = (c00 > 0.0f) ? (1.0f / fmaxf(c00, 1.0f)) : 0.0f;
    const float scl1  = (c01 > 0.0f) ? (1.0f / fmaxf(c01, 1.0f)) : 0.0f;

    const float* aMean0 = agg_sg + (size_t)ar0c * DIM;
    const float* aMean1 = agg_sg + (size_t)ar1c * DIM;
    const float* aX0    = xin_sg + (size_t)ar0c * DIM;
    const float* aX1    = xin_sg + (size_t)ar1c * DIM;

    v8f acc0[4], acc1[4];
    #pragma unroll
    for (int t = 0; t < 4; ++t) {
        const float bv = sB[t * 16 + lo];
        #pragma unroll
        for (int r = 0; r < 8; ++r) { acc0[t][r] = bv; acc1[t][r] = bv; }
    }

    // mean @ Wl  (B-fragments shared across both row tiles)
    for (int kk = 0; kk < 16; ++kk) {
        const int kb = kk * 4 + hi * 2;
        const v2f am0 = *(const v2f*)(aMean0 + kb);
        const v2f am1 = *(const v2f*)(aMean1 + kb);
        v2f a0; a0.x = am0.x * scl0; a0.y = am0.y * scl0;
        v2f a1; a1.x = am1.x * scl1; a1.y = am1.y * scl1;
        #pragma unroll
        for (int t = 0; t < 4; ++t) {
            v2f b;
            b.x = sWl[(kb    ) * DIM + t * 16 + lo];
            b.y = sWl[(kb + 1) * DIM + t * 16 + lo];
            acc0[t] = wmma_f32(a0, b, acc0[t]);
            acc1[t] = wmma_f32(a1, b, acc1[t]);
        }
    }
    // x @ Wr
    for (int kk = 0; kk < 16; ++kk) {
        const int kb = kk * 4 + hi * 2;
        const v2f a0 = *(const v2f*)(aX0 + kb);
        const v2f a1 = *(const v2f*)(aX1 + kb);
        #pragma unroll
        for (int t = 0; t < 4; ++t) {
            v2f b;
            b.x = sWr[(kb    ) * DIM + t * 16 + lo];
            b.y = sWr[(kb + 1) * DIM + t * 16 + lo];
            acc0[t] = wmma_f32(a0, b, acc0[t]);
            acc1[t] = wmma_f32(a1, b, acc1[t]);
        }
    }

    #pragma unroll
    for (int t = 0; t < 4; ++t) {
        #pragma unroll
        for (int r = 0; r < 8; ++r) {
            const int rr0 = row0 + r + 8 * hi;
            const int rr1 = row0 + 16 + r + 8 * hi;
            if (rr0 < NNODES)
                out_sg[(size_t)rr0 * DIM + t * 16 + lo] = tanhf(acc0[t][r]);
            if (rr1 < NNODES)
                out_sg[(size_t)rr1 * DIM + t * 16 + lo] = tanhf(acc1[t][r]);
        }
    }
}

// ---------------------------------------------------------------------------
// 3) LSTM over T=G=3 steps; batch B = S*N = 80000; hidden 64.
//    xt for step t is h2[s][g=t][n]. h/c live in registers (C-layout tiles);
//    h is staged through per-wave LDS between steps to re-read in A-layout.
//    gates: 16 accumulator tiles (i,f,g,o x 4 col tiles), fp32 WMMA.
//    Wih/Whh (128 KB total) stay hot in L2/L0 -> read B-fragments directly.
// ---------------------------------------------------------------------------
__global__ void lstm_kernel(const float* __restrict__ h2,   // [S][G][N][64]
                            const float* __restrict__ Wih,  // [256][64]
                            const float* __restrict__ Whh,  // [256][64]
                            const float* __restrict__ bih,  // [256]
                            const float* __restrict__ bhh,  // [256]
                            float* __restrict__ out)        // [B][64]
{
    __shared__ float sH[8][16 * DIM];   // 32 KB: per-wave h staging

    const int lane = threadIdx.x & 31;
    const int wave = threadIdx.x >> 5;
    const int lo   = lane & 15;
    const int hi   = lane >> 4;
    const int b0   = (blockIdx.x * 8 + wave) * 16;   // B divides exactly

    float* myH = sH[wave];

    v8f hC[4], cC[4];
    #pragma unroll
    for (int c = 0; c < 4; ++c)
        #pragma unroll
        for (int r = 0; r < 8; ++r) { hC[c][r] = 0.0f; cC[c][r] = 0.0f; }

    // A-fragment row for this lane
    const int b = b0 + lo;
    const int s = b / NNODES;
    const int n = b - s * NNODES;

    for (int t = 0; t < GNUM; ++t) {
        // stage previous h (C-layout) into LDS, re-read in A-layout
        #pragma unroll
        for (int c = 0; c < 4; ++c)
            #pragma unroll
            for (int r = 0; r < 8; ++r)
                myH[(r + 8 * hi) * DIM + c * 16 + lo] = hC[c][r];
        __syncthreads();

        v8f gAcc[16];
        #pragma unroll
        for (int j = 0; j < 16; ++j) {
            const float bv = bih[j * 16 + lo] + bhh[j * 16 + lo];
            #pragma unroll
            for (int r = 0; r < 8; ++r) gAcc[j][r] = bv;
        }

        const float* xrow = h2 + (((size_t)s * GNUM + t) * NNODES + n) * DIM;

        // gates += xt @ Wih^T   (B[k][j] = Wih[j][k])
        for (int kk = 0; kk < 16; ++kk) {
            const int kb = kk * 4 + hi * 2;
            const v2f a = *(const v2f*)(xrow + kb);
            #pragma unroll
            for (int j = 0; j < 16; ++j) {
                const v2f bb = *(const v2f*)(Wih + (size_t)(j * 16 + lo) * DIM + kb);
                gAcc[j] = wmma_f32(a, bb, gAcc[j]);
            }
        }
        // gates += h_prev @ Whh^T
        for (int kk = 0; kk < 16; ++kk) {
            const int kb = kk * 4 + hi * 2;
            v2f a;
            a.x = myH[lo * DIM + kb];
            a.y = myH[lo * DIM + kb + 1];
            #pragma unroll
            for (int j = 0; j < 16; ++j) {
                const v2f bb = *(const v2f*)(Whh + (size_t)(j * 16 + lo) * DIM + kb);
                gAcc[j] = wmma_f32(a, bb, gAcc[j]);
            }
        }

        // elementwise cell update (torch gate order i,f,g,o)
        #pragma unroll
        for (int c = 0; c < 4; ++c)
            #pragma unroll
            for (int r = 0; r < 8; ++r) {
                const float ig = sigmoidf_(gAcc[c     ][r]);
                const float fg = sigmoidf_(gAcc[c + 4 ][r]);
                const float gg = tanhf   (gAcc[c + 8 ][r]);
                const float og = sigmoidf_(gAcc[c + 12][r]);
                const float cn = fg * cC[c][r] + ig * gg;
                cC[c][r] = cn;
                hC[c][r] = og * tanhf(cn);
            }
        __syncthreads();
    }

    #pragma unroll
    for (int c = 0; c < 4; ++c)
        #pragma unroll
        for (int r = 0; r < 8; ++r)
            out[(size_t)(b0 + r + 8 * hi) * DIM + c * 16 + lo] = hC[c][r];
}

// ---------------------------------------------------------------------------
// launcher
// ---------------------------------------------------------------------------
extern "C" void kernel_launch(void* const* d_in, const int* in_sizes, int n_in,
                              void* d_out, int out_size, void* d_ws, size_t ws_size,
                              hipStream_t stream) {
    const float* x    = (const float*)d_in[0];   // [S,G,N,64]
    const int*   ei   = (const int*)  d_in[1];   // [S,G,2,E]
    const float* W1l  = (const float*)d_in[2];
    const float* W1r  = (const float*)d_in[3];
    const float* b1   = (const float*)d_in[4];
    const float* W2l  = (const float*)d_in[5];
    const float* W2r  = (const float*)d_in[6];
    const float* b2   = (const float*)d_in[7];
    const float* Wih  = (const float*)d_in[8];
    const float* Whh  = (const float*)d_in[9];
    const float* bih  = (const float*)d_in[10];
    const float* bhh  = (const float*)d_in[11];
    float*       out  = (float*)d_out;

    // workspace: agg | cnt | h1 | h2
    float* agg = (float*)d_ws;
    float* cnt = agg + (size_t)SG * NNODES * DIM;
    float* h1  = cnt + (size_t)SG * NNODES;
    float* h2  = h1  + (size_t)SG * NNODES * DIM;

    const long long nz = (long long)SG * NNODES * DIM + (long long)SG * NNODES;
    const int zb = (int)((nz + 255) / 256);

    dim3 blk(256);
    dim3 gScatter(EDGES / 8, SG);                      // 8 edges / 256-thr block
    dim3 gSage((NNODES + 255) / 256, SG);              // 8 waves x 32 rows
    dim3 gLstm(BATCH / 128);                           // 80000 / (8 waves * 16)

    // ---- layer 1 ----
    zero_kernel<<<zb, blk, 0, stream>>>(agg, nz);
    scatter_kernel<<<gScatter, blk, 0, stream>>>(x, ei, agg, cnt);
    sage_linear_kernel<<<gSage, blk, 0, stream>>>(x, agg, cnt, W1l, W1r, b1, h1);

    // ---- layer 2 ----
    zero_kernel<<<zb, blk, 0, stream>>>(agg, nz);
    scatter_kernel<<<gScatter, blk, 0, stream>>>(h1, ei, agg, cnt);
    sage_linear_kernel<<<gSage, blk, 0, stream>>>(h1, agg, cnt, W2l, W2r, b2, h2);

    // ---- LSTM ----
    lstm_kernel<<<gLstm, blk, 0, stream>>>(h2, Wih, Whh, bih, bhh, out);
}